// CURLoRAModule_34325378629967
// MI455X (gfx1250) — compile-verified
//
#include <hip/hip_runtime.h>

// CUR low-rank forward:  out[p,n] = sum_r (x @ R^T @ U^T)[p,r] * C[n,r]
//   x: [128, 8192] f32, C: [8192, 64] f32, U: [64,64] f32, R: [64, 8192] f32
// Fused low-rank chain (never materialize W). Memory-bound (~12 MB total),
// fp32 WMMA (v_wmma_f32_16x16x4_f32) for all three GEMMs.

typedef __attribute__((ext_vector_type(2))) float v2f;
typedef __attribute__((ext_vector_type(8))) float v8f;

#define M_DIM   8192
#define N_DIM   8192
#define RANK    64
#define P_DIM   128                    // B*S = 2*64
#define S1_BLOCKS 16                   // stage-1 workgroups (M-chunks)
#define S1_CHUNK  (M_DIM / S1_BLOCKS)  // 512

static __device__ __forceinline__ v8f wmma_f32_16x16x4(v2f a, v2f b, v8f c) {
  // 8 args: (neg_a, A, neg_b, B, c_mod, C, reuse_a, reuse_b)
  return __builtin_amdgcn_wmma_f32_16x16x4_f32(false, a, false, b,
                                               (short)0, c, false, false);
}

// ---------------------------------------------------------------------------
// Stage 1: partial[c][p][r] = sum_{m in chunk c} x[p][m] * R[r][m]
// Grid: S1_BLOCKS x 256 threads (8 waves). Wave w owns tiles w, w+8, w+16, w+24
// of the 8x4 grid of 16x16 tiles covering t1 [128 x 64].
// ---------------------------------------------------------------------------
__global__ void __launch_bounds__(256)
cur_stage1_xRt(const float* __restrict__ x, const float* __restrict__ R,
               float* __restrict__ partial) {
  const int lane = threadIdx.x & 31;
  const int wave = threadIdx.x >> 5;
  const int half = lane >> 4;        // 0: K={0,1} side, 1: K={2,3} side
  const int l16  = lane & 15;
  const int kk   = half * 2;
  const int mBase = blockIdx.x * S1_CHUNK;

  float* dst = partial + (size_t)blockIdx.x * (P_DIM * RANK);

  for (int t = wave; t < 32; t += 8) {
    const int pBase = (t & 7) * 16;
    const int rBase = (t >> 3) * 16;
    // A lane source: row of x (M = l16); B lane source: row of R (N = l16)
    const float* xRow = x + (size_t)(pBase + l16) * M_DIM;
    const float* rRow = R + (size_t)(rBase + l16) * M_DIM;

    v8f acc = {};
    for (int m = mBase; m < mBase + S1_CHUNK; m += 4) {
      __builtin_prefetch(xRow + m + 128, 0, 0);   // speculative stream hint
      v2f a = *reinterpret_cast<const v2f*>(xRow + m + kk);
      v2f b = *reinterpret_cast<const v2f*>(rRow + m + kk);
      acc = wmma_f32_16x16x4(a, b, acc);
    }

    // D layout: VGPR v -> row (v + 8*half), col l16
#pragma unroll
    for (int v = 0; v < 8; ++v)
      dst[(pBase + v + 8 * half) * RANK + (rBase + l16)] = acc[v];
  }
}

// ---------------------------------------------------------------------------
// Stage 2: t1 = sum_c partial[c] (fixed order -> deterministic), then
//          t2[p][u] = sum_r t1[p][r] * U[u][r].  Single workgroup, t1 in LDS.
// ---------------------------------------------------------------------------
__global__ void __launch_bounds__(256)
cur_stage2_Ut(const float* __restrict__ partial, const float* __restrict__ U,
              float* __restrict__ t2) {
  __shared__ float t1s[P_DIM * RANK];   // 32 KB of the 320 KB WGP LDS

  for (int e = threadIdx.x; e < P_DIM * RANK; e += 256) {
    float s = 0.f;
#pragma unroll
    for (int c = 0; c < S1_BLOCKS; ++c)
      s += partial[c * (P_DIM * RANK) + e];
    t1s[e] = s;
  }
  __syncthreads();

  const int lane = threadIdx.x & 31;
  const int wave = threadIdx.x >> 5;
  const int half = lane >> 4;
  const int l16  = lane & 15;
  const int kk   = half * 2;

  for (int t = wave; t < 32; t += 8) {
    const int pBase = (t & 7) * 16;
    const int uBase = (t >> 3) * 16;
    const float* uRow = U + (size_t)(uBase + l16) * RANK;

    v8f acc = {};
#pragma unroll
    for (int r0 = 0; r0 < RANK; r0 += 4) {
      v2f a = *reinterpret_cast<const v2f*>(&t1s[(pBase + l16) * RANK + r0 + kk]);
      v2f b = *reinterpret_cast<const v2f*>(uRow + r0 + kk);
      acc = wmma_f32_16x16x4(a, b, acc);
    }
#pragma unroll
    for (int v = 0; v < 8; ++v)
      t2[(pBase + v + 8 * half) * RANK + (uBase + l16)] = acc[v];
  }
}

// ---------------------------------------------------------------------------
// Stage 3: out[p][n] = sum_r t2[p][r] * C[n][r].  4096 16x16 tiles, one per
// wave; 512 blocks x 8 waves. K = 64 -> 16 fully-unrolled WMMA steps.
// ---------------------------------------------------------------------------
__global__ void __launch_bounds__(256)
cur_stage3_Ct(const float* __restrict__ t2, const float* __restrict__ C,
              float* __restrict__ out) {
  const int lane = threadIdx.x & 31;
  const int wave = threadIdx.x >> 5;
  const int half = lane >> 4;
  const int l16  = lane & 15;
  const int kk   = half * 2;

  const int tile  = blockIdx.x * 8 + wave;   // 0..4095
  const int pBase = (tile & 7) * 16;
  const int nBase = (tile >> 3) * 16;

  const float* aRow = t2 + (size_t)(pBase + l16) * RANK;
  const float* cRow = C  + (size_t)(nBase + l16) * RANK;

  v8f acc = {};
#pragma unroll
  for (int r0 = 0; r0 < RANK; r0 += 4) {
    v2f a = *reinterpret_cast<const v2f*>(aRow + r0 + kk);
    v2f b = *reinterpret_cast<const v2f*>(cRow + r0 + kk);
    acc = wmma_f32_16x16x4(a, b, acc);
  }

#pragma unroll
  for (int v = 0; v < 8; ++v)
    out[(size_t)(pBase + v + 8 * half) * N_DIM + (nBase + l16)] = acc[v];
}

// ---------------------------------------------------------------------------
extern "C" void kernel_launch(void* const* d_in, const int* in_sizes, int n_in,
                              void* d_out, int out_size, void* d_ws, size_t ws_size,
                              hipStream_t stream) {
  (void)in_sizes; (void)n_in; (void)out_size; (void)ws_size;
  const float* x = (const float*)d_in[0];   // [128, 8192]
  const float* C = (const float*)d_in[1];   // [8192, 64]
  const float* U = (const float*)d_in[2];   // [64, 64]
  const float* R = (const float*)d_in[3];   // [64, 8192]
  float* out = (float*)d_out;               // [128, 8192]

  float* partial = (float*)d_ws;                          // 16*128*64 f32 = 512 KB
  float* t2      = partial + S1_BLOCKS * P_DIM * RANK;    // 128*64 f32

  cur_stage1_xRt<<<S1_BLOCKS, 256, 0, stream>>>(x, R, partial);
  cur_stage2_Ut <<<1,         256, 0, stream>>>(partial, U, t2);
  cur_stage3_Ct <<<(4096 / 8), 256, 0, stream>>>(t2, C, out);
}